// MambaLayer_47364899340729
// MI455X (gfx1250) — compile-verified
//
#include <hip/hip_runtime.h>
#include <hip/hip_bf16.h>
#include <math.h>

// ---------------------------------------------------------------------------
// Mamba layer for MI455X (gfx1250, wave32, WMMA + async-LDS staging).
#define D_MODEL 1024
#define D_STATE 16
#define D_CONV  4
#define D_INNER 2048
#define DT_RANK 64
#define B_SZ    4
#define SEQ     2048
#define MROWS   (B_SZ * SEQ)          // 8192
#define RMS_EPS 1e-12f
// ---------------------------------------------------------------------------

typedef float v2f __attribute__((ext_vector_type(2)));
typedef float v8f __attribute__((ext_vector_type(8)));

__device__ __forceinline__ float silu_f(float x) {
    return x / (1.0f + __expf(-x));
}

// ---------------------------------------------------------------------------
// CDNA5 async global->LDS copy (ASYNCcnt-tracked). Inline asm is portable
// across ROCm 7.2 and the amdgpu-toolchain (bypasses clang builtin arity
// differences). First operand = VGPR holding the 32-bit LDS byte address
// (low 32 bits of the generic address), second = 64-bit global address.
// ---------------------------------------------------------------------------
__device__ __forceinline__ void async_copy_b128(const float* g, const float* lds) {
    unsigned l32 = (unsigned)(uintptr_t)lds;       // generic low 32 bits == LDS offset
    unsigned long long g64 = (unsigned long long)(uintptr_t)g;
    asm volatile("global_load_async_to_lds_b128 %0, %1, off"
                 :: "v"(l32), "v"(g64)
                 : "memory");
}

#define WAIT_ASYNCCNT(n) asm volatile("s_wait_asynccnt " #n ::: "memory")

// ===========================================================================
// fp32 WMMA GEMM with double-buffered async-LDS staging.
//   C[M,N] = act( X[M,ldx(>=K)] * W[N,K]^T + bias )
//   block = 128 threads = 4 waves; tile = 64 rows x 64 cols; K-chunk = 32.
//   B tile is shared by all 4 waves (staged once per block).
//   EPI: 0 = identity(+bias), 1 = softplus(+bias), 2 = exact GELU(+bias)
// ===========================================================================
#define KC   32
#define KCP  36   // padded LDS row stride (floats); 144 B keeps 16-B alignment

template <int EPI>
__global__ __launch_bounds__(128)
void gemm_wmma_kernel(const float* __restrict__ X, int ldx,
                      const float* __restrict__ W,
                      const float* __restrict__ bias,
                      float* __restrict__ C,
                      int M, int N, int K)
{
    __shared__ float As[2][64][KCP];
    __shared__ float Bs[2][64][KCP];

    const int tid  = threadIdx.x;
    const int lane = tid & 31;
    const int wave = tid >> 5;
    const int lm   = lane & 15;
    const int half = lane >> 4;

    const int m0 = blockIdx.y * 64;                // block row base
    const int n0 = blockIdx.x * 64;                // block col base

    // Stage one K-chunk (A: 64x32 floats, B: 64x32 floats) with async b128
    // copies: 8 per thread per chunk (4 A + 4 B) -> ASYNCcnt 8 per wave.
    auto stage = [&](int k0, int buf) {
#pragma unroll
        for (int i = 0; i < 4; ++i) {
            int li  = tid + i * 128;               // 0..511
            int row = li >> 3;                     // 0..63
            int q   = (li & 7) * 4;                // float offset 0,4,..,28
            async_copy_b128(X + (size_t)(m0 + row) * (size_t)ldx + k0 + q,
                            &As[buf][row][q]);
            int n = n0 + row;
            if (n >= N) n = N - 1;                 // clamp (stores are guarded)
            async_copy_b128(W + (size_t)n * (size_t)K + k0 + q,
                            &Bs[buf][row][q]);
        }
    };

    v8f acc[4];
#pragma unroll
    for (int t = 0; t < 4; ++t)
#pragma unroll
        for (int v = 0; v < 8; ++v) acc[t][v] = 0.0f;

    const int nch = K / KC;

    stage(0, 0);                                    // prologue: chunk 0

    for (int c = 0; c < nch; ++c) {
        const int buf = c & 1;
        if (c + 1 < nch) {
            stage((c + 1) * KC, buf ^ 1);           // prefetch next chunk
            WAIT_ASYNCCNT(8);                       // chunk c landed, c+1 in flight
        } else {
            WAIT_ASYNCCNT(0);                       // drain
        }
        __syncthreads();                            // all waves' data visible

        // 8 K-steps x 4 WMMAs from LDS (ds_load_b64 fragments).
        const float* aRow = &As[buf][wave * 16 + lm][2 * half];
#pragma unroll
        for (int kk = 0; kk < KC / 4; ++kk) {
            int k = kk * 4;
            v2f a = *(const v2f*)(aRow + k);
#pragma unroll
            for (int t = 0; t < 4; ++t) {
                v2f b = *(const v2f*)(&Bs[buf][t * 16 + lm][2 * half + k]);
                acc[t] = __builtin_amdgcn_wmma_f32_16x16x4_f32(
                    false, a, false, b, (short)0, acc[t], false, false);
            }
        }
        __syncthreads();                            // done reading buf before reuse
    }

    // Epilogue. C layout: VGPR v <-> row m0+wave*16 + v + 8*half, col n0+t*16+lm.
#pragma unroll
    for (int t = 0; t < 4; ++t) {
        int n = n0 + t * 16 + lm;
        if (n >= N) continue;
        float bsv = bias ? bias[n] : 0.0f;
#pragma unroll
        for (int v = 0; v < 8; ++v) {
            int m = m0 + wave * 16 + v + half * 8;
            float val = acc[t][v] + bsv;
            if (EPI == 1) {        // softplus
                val = (val > 20.0f) ? val : log1pf(__expf(val));
            } else if (EPI == 2) { // exact GELU
                val = 0.5f * val * (1.0f + erff(val * 0.70710678118654752f));
            }
            C[(size_t)m * (size_t)N + n] = val;
        }
    }
}

// ===========================================================================
// Causal depthwise conv (width 4) + SiLU.
// ===========================================================================
__global__ void conv_silu_kernel(const float* __restrict__ xz,
                                 const float* __restrict__ cw,
                                 const float* __restrict__ cb,
                                 float* __restrict__ u)
{
    size_t idx = (size_t)blockIdx.x * blockDim.x + threadIdx.x;
    const size_t total = (size_t)MROWS * D_INNER;
    if (idx >= total) return;
    int d   = (int)(idx % D_INNER);
    int row = (int)(idx / D_INNER);          // row = b*SEQ + t
    int t   = row % SEQ;
    int b   = row / SEQ;

    const float* xcol = xz + (size_t)b * SEQ * (2 * D_INNER) + d;
    float acc = cb[d];
#pragma unroll
    for (int j = 0; j < D_CONV; ++j) {
        int tt = t - (D_CONV - 1) + j;
        if (tt >= 0) acc += xcol[(size_t)tt * (2 * D_INNER)] * cw[d * D_CONV + j];
    }
    u[idx] = silu_f(acc);
}

// ===========================================================================
// Selective scan: one thread per (b,d); 16-wide state in registers;
// B_t/C_t staged through LDS. Fuses +u*D and *silu(z).
// ===========================================================================
__global__ __launch_bounds__(256)
void scan_kernel(const float* __restrict__ delta,
                 const float* __restrict__ u,
                 const float* __restrict__ xdbl,
                 const float* __restrict__ xz,
                 const float* __restrict__ A_log,
                 const float* __restrict__ Dp,
                 float* __restrict__ ys)
{
    const int chunks = D_INNER / 256;                // 8
    int b = blockIdx.x / chunks;
    int c = blockIdx.x % chunks;
    int d = c * 256 + threadIdx.x;

    const float* dl = delta + (size_t)b * SEQ * D_INNER + d;
    const float* up = u     + (size_t)b * SEQ * D_INNER + d;
    const float* zp = xz    + (size_t)b * SEQ * (2 * D_INNER) + D_INNER + d;
    const float* xb = xdbl  + (size_t)b * SEQ * 96;
    float*       yp = ys    + (size_t)b * SEQ * D_INNER + d;

    float A[D_STATE];
#pragma unroll
    for (int n = 0; n < D_STATE; ++n) A[n] = -__expf(A_log[(size_t)d * D_STATE + n]);
    const float Dd = Dp[d];

    float h[D_STATE];
#pragma unroll
    for (int n = 0; n < D_STATE; ++n) h[n] = 0.0f;

    __shared__ float sBC[32];

    for (int t = 0; t < SEQ; ++t) {
        if (threadIdx.x < 32)
            sBC[threadIdx.x] = xb[(size_t)t * 96 + DT_RANK + threadIdx.x];
        __syncthreads();

        float dv = dl[(size_t)t * D_INNER];
        float uv = up[(size_t)t * D_INNER];
        float du = dv * uv;
        float y = 0.0f;
#pragma unroll
        for (int n = 0; n < D_STATE; ++n) {
            float dA = __expf(dv * A[n]);
            h[n] = dA * h[n] + du * sBC[n];
            y += h[n] * sBC[16 + n];
        }
        float zv = zp[(size_t)t * (2 * D_INNER)];
        yp[(size_t)t * D_INNER] = (y + uv * Dd) * silu_f(zv);
        __syncthreads();
    }
}

// ===========================================================================
// RMSNorm over last dim (1024): out = (x+res)*rsqrt(mean((x+res)^2)+eps)*w
// ===========================================================================
__global__ __launch_bounds__(256)
void rmsnorm_kernel(const float* __restrict__ x,
                    const float* __restrict__ res,
                    const float* __restrict__ w,
                    float* __restrict__ out)
{
    const int tid = threadIdx.x;
    const size_t base = (size_t)blockIdx.x * D_MODEL;

    float vals[4];
    float s = 0.0f;
#pragma unroll
    for (int i = 0; i < 4; ++i) {
        int j = tid + i * 256;
        float v = x[base + j] + res[base + j];
        vals[i] = v;
        s += v * v;
    }

    __shared__ float red[256];
    red[tid] = s;
    __syncthreads();
    for (int off = 128; off > 0; off >>= 1) {
        if (tid < off) red[tid] += red[tid + off];
        __syncthreads();
    }
    float scale = rsqrtf(red[0] * (1.0f / D_MODEL) + RMS_EPS);

#pragma unroll
    for (int i = 0; i < 4; ++i) {
        int j = tid + i * 256;
        out[base + j] = vals[i] * scale * w[j];
    }
}

// ===========================================================================
// Host launcher
// ===========================================================================
extern "C" void kernel_launch(void* const* d_in, const int* in_sizes, int n_in,
                              void* d_out, int out_size, void* d_ws, size_t ws_size,
                              hipStream_t stream)
{
    const float* x_in      = (const float*)d_in[0];
    const float* in_proj_w = (const float*)d_in[1];
    const float* conv_w    = (const float*)d_in[2];
    const float* conv_b    = (const float*)d_in[3];
    const float* x_proj_w  = (const float*)d_in[4];
    const float* dt_proj_w = (const float*)d_in[5];
    const float* dt_proj_b = (const float*)d_in[6];
    const float* A_log     = (const float*)d_in[7];
    const float* Dp        = (const float*)d_in[8];
    const float* out_proj_w= (const float*)d_in[9];
    const float* norm_w    = (const float*)d_in[10];
    const float* ffn_w1    = (const float*)d_in[11];
    const float* ffn_b1    = (const float*)d_in[12];
    const float* ffn_w2    = (const float*)d_in[13];
    const float* ffn_b2    = (const float*)d_in[14];
    const float* ffn_norm_w= (const float*)d_in[15];
    float* out = (float*)d_out;

    float* ws = (float*)d_ws;
    const size_t SZ_XZ   = (size_t)MROWS * 2 * D_INNER;
    const size_t SZ_U    = (size_t)MROWS * D_INNER;
    const size_t SZ_XDBL = (size_t)MROWS * 96;
    const size_t SZ_DLT  = (size_t)MROWS * D_INNER;
    const size_t SZ_YS   = (size_t)MROWS * D_INNER;
    const size_t SZ_MO   = (size_t)MROWS * D_MODEL;

    float* xz    = ws;                       // [MROWS,4096]; reused as ffn_mid
    float* u     = xz    + SZ_XZ;            // [MROWS,2048]
    float* xdbl  = u     + SZ_U;             // [MROWS,96]
    float* delta = xdbl  + SZ_XDBL;          // [MROWS,2048]
    float* ysb   = delta + SZ_DLT;           // [MROWS,2048]
    float* mo    = ysb   + SZ_YS;            // [MROWS,1024]; reused as ffn out
    float* hbuf  = mo    + SZ_MO;            // [MROWS,1024]

    const dim3 gblk(128);

    // 1) xz = x @ in_proj_w^T           (N=4096, K=1024)
    gemm_wmma_kernel<0><<<dim3(4096 / 64, MROWS / 64), gblk, 0, stream>>>(
        x_in, D_MODEL, in_proj_w, nullptr, xz, MROWS, 2 * D_INNER, D_MODEL);

    // 2) u = silu(causal_dwconv(xz[:, :D_INNER]))
    {
        size_t total = (size_t)MROWS * D_INNER;
        conv_silu_kernel<<<(unsigned)((total + 255) / 256), 256, 0, stream>>>(
            xz, conv_w, conv_b, u);
    }

    // 3) x_dbl = u @ x_proj_w^T         (N=96, K=2048)
    gemm_wmma_kernel<0><<<dim3((96 + 63) / 64, MROWS / 64), gblk, 0, stream>>>(
        u, D_INNER, x_proj_w, nullptr, xdbl, MROWS, 96, D_INNER);

    // 4) delta = softplus(x_dbl[:, :64] @ dt_proj_w^T + dt_proj_b)  (N=2048, K=64)
    gemm_wmma_kernel<1><<<dim3(2048 / 64, MROWS / 64), gblk, 0, stream>>>(
        xdbl, 96, dt_proj_w, dt_proj_b, delta, MROWS, D_INNER, DT_RANK);

    // 5) selective scan (+ u*D, * silu(z))
    scan_kernel<<<B_SZ * (D_INNER / 256), 256, 0, stream>>>(
        delta, u, xdbl, xz, A_log, Dp, ysb);

    // 6) mo = ys @ out_proj_w^T         (N=1024, K=2048)
    gemm_wmma_kernel<0><<<dim3(1024 / 64, MROWS / 64), gblk, 0, stream>>>(
        ysb, D_INNER, out_proj_w, nullptr, mo, MROWS, D_MODEL, D_INNER);

    // 7) h = rmsnorm(mo + x, norm_w)
    rmsnorm_kernel<<<MROWS, 256, 0, stream>>>(mo, x_in, norm_w, hbuf);

    // 8) ffn_mid = gelu(h @ ffn_w1^T + b1)   (reuses xz; N=4096, K=1024)
    gemm_wmma_kernel<2><<<dim3(4096 / 64, MROWS / 64), gblk, 0, stream>>>(
        hbuf, D_MODEL, ffn_w1, ffn_b1, xz, MROWS, 4 * D_MODEL, D_MODEL);

    // 9) ff = ffn_mid @ ffn_w2^T + b2        (reuses mo; N=1024, K=4096)
    gemm_wmma_kernel<0><<<dim3(1024 / 64, MROWS / 64), gblk, 0, stream>>>(
        xz, 4 * D_MODEL, ffn_w2, ffn_b2, mo, MROWS, D_MODEL, 4 * D_MODEL);

    // 10) out = rmsnorm(ff + h, ffn_norm_w)
    rmsnorm_kernel<<<MROWS, 256, 0, stream>>>(mo, hbuf, ffn_norm_w, out);
}